// NeuralNet_67602785239231
// MI455X (gfx1250) — compile-verified
//
#include <hip/hip_runtime.h>
#include <hip/hip_bf16.h>
#include <math.h>

// Problem constants from the reference
#define NN   512
#define CC   7
#define DM   300
#define DMP  304              // DM rounded up to multiple of 16 (row padding)
#define KK   3
#define NC   (NN*CC)          // 3584
#define LBP  10

typedef __attribute__((ext_vector_type(2))) float v2f;
typedef __attribute__((ext_vector_type(8))) float v8f;

__device__ __forceinline__ v8f wmma4(v2f a, v2f b, v8f c) {
  // D = A(16x4,f32) * B(4x16,f32) + C(16x16,f32)
  return __builtin_amdgcn_wmma_f32_16x16x4_f32(
      /*neg_a=*/false, a, /*neg_b=*/false, b,
      /*c_mod=*/(short)0, c, /*reuse_a=*/false, /*reuse_b=*/false);
}

__device__ __forceinline__ v2f ldv2(const float* p) {
  return *(const v2f*)p;      // 8-byte aligned by construction
}

// ---------------------------------------------------------------------------
// C = A (MxK, lda) * B^T (B is NpadxK rows, ldb), one wave per 16x16 tile.
// REQUIREMENTS: M % 16 == 0; B has >= gridDim.x*16 valid (possibly zero) rows;
// K % 4 == 0; rows 8-byte aligned. No predication in the K loop.
// epilogue: mode 0 -> plain, 1 -> tanh(acc + bias[n]), 2 -> exp(acc*scale)
// store:   C[m*ldc + n*cstride + coff], guarded by n < N.
// ---------------------------------------------------------------------------
__global__ void gemm_nt_wmma(const float* __restrict__ A, const float* __restrict__ B,
                             float* __restrict__ C, const float* __restrict__ bias,
                             int M, int N, int K, int lda, int ldb, int ldc,
                             int cstride, int coff, int mode, float scale)
{
  const int lane = threadIdx.x;          // block = 1 wave32
  const int half = lane >> 4;            // 0: K={0,1}, 1: K={2,3}
  const int l16  = lane & 15;
  const int klo  = half * 2;

  const int tileM = blockIdx.y, tileN = blockIdx.x;
  const float* __restrict__ arow = A + (size_t)(tileM * 16 + l16) * lda + klo;
  const float* __restrict__ brow = B + (size_t)(tileN * 16 + l16) * ldb + klo;

  v8f acc = {};
#pragma unroll 4
  for (int k = 0; k < K; k += 4) {
    v2f a = ldv2(arow + k);
    v2f b = ldv2(brow + k);
    acc = wmma4(a, b, acc);
  }

  // C/D layout: lanes 0-15 hold M=r (vgpr r), lanes 16-31 hold M=r+8; N = lane&15
  const int mbase = tileM * 16 + half * 8;
  const int nn = tileN * 16 + l16;
  if (nn < N) {
#pragma unroll
    for (int r = 0; r < 8; ++r) {
      int mm = mbase + r;
      float v = acc[r];
      if (mode == 1)      v = tanhf(v + bias[nn]);
      else if (mode == 2) v = __expf(v * scale);
      C[(size_t)mm * ldc + (size_t)nn * cstride + coff] = v;
    }
  }
}

// ---------------------------------------------------------------------------
// phis2[ic][jp] = sum_k ass[i,j,k] * sum_d val_k[ic,d]*emb[jp,d]
// (layout (i,c) x (j,p); the LBP kernel indexes it accordingly)
// One wave per 16x16 tile, 3 accumulators (one per k), fused ass-combine.
// All dims are multiples of 16 -> no predication anywhere.
// ---------------------------------------------------------------------------
__global__ void phis_wmma(const float* __restrict__ val, const float* __restrict__ emb,
                          const float* __restrict__ ass, float* __restrict__ phis)
{
  const int lane = threadIdx.x;
  const int half = lane >> 4;
  const int l16  = lane & 15;
  const int klo  = half * 2;
  const int tileM = blockIdx.y, tileN = blockIdx.x;

  const size_t vstride = (size_t)NC * DM;
  const float* __restrict__ a0 = val + (size_t)(tileM * 16 + l16) * DM + klo;
  const float* __restrict__ a1 = a0 + vstride;
  const float* __restrict__ a2 = a1 + vstride;
  const float* __restrict__ br = emb + (size_t)(tileN * 16 + l16) * DM + klo;

  v8f c0 = {}, c1 = {}, c2 = {};
#pragma unroll 2
  for (int k = 0; k < DM; k += 4) {
    v2f b  = ldv2(br + k);
    v2f x0 = ldv2(a0 + k);
    v2f x1 = ldv2(a1 + k);
    v2f x2 = ldv2(a2 + k);
    c0 = wmma4(x0, b, c0);
    c1 = wmma4(x1, b, c1);
    c2 = wmma4(x2, b, c2);
  }

  const int mbase = tileM * 16 + half * 8;
  const int nn = tileN * 16 + l16;
  const int j = nn / CC;
#pragma unroll
  for (int r = 0; r < 8; ++r) {
    int mm = mbase + r;
    int i = mm / CC;
    const float* aw = ass + ((size_t)i * NN + j) * KK;
    phis[(size_t)mm * NC + nn] = c0[r] * aw[0] + c1[r] * aw[1] + c2[r] * aw[2];
  }
}

// ---------------------------------------------------------------------------
// Small elementwise / reduction kernels
// ---------------------------------------------------------------------------

// Copy src (rows x cols) into dst (rowsPad x cols), zero-filling padded rows.
__global__ void pad_rows(const float* __restrict__ src, float* __restrict__ dst,
                         int rows, int cols, int rowsPad)
{
  int idx = blockIdx.x * blockDim.x + threadIdx.x;
  if (idx >= rowsPad * cols) return;
  int r = idx / cols;
  int c = idx - r * cols;
  dst[idx] = (r < rows) ? src[(size_t)r * cols + c] : 0.0f;
}

// Rtp[k][d][e] = R[k][e][d] for d < DM, zero rows for d in [DM, DMP)
__global__ void transpose_pad_R(const float* __restrict__ R, float* __restrict__ Rt)
{
  int idx = blockIdx.x * blockDim.x + threadIdx.x;   // k*DMP*DM + d*DM + e
  if (idx >= KK * DMP * DM) return;
  int k = idx / (DMP * DM);
  int r = idx - k * DMP * DM;
  int d = r / DM, e = r - d * DM;
  Rt[idx] = (d < DM) ? R[(size_t)k * DM * DM + (size_t)e * DM + d] : 0.0f;
}

__global__ void psiss_k(const float* __restrict__ emb, const float* __restrict__ Bf,
                        float* __restrict__ psiss)
{
  int idx = blockIdx.x * blockDim.x + threadIdx.x;    // n*7 + c
  if (idx >= NC) return;
  int n = idx / CC;
  const float* e  = emb + (size_t)idx * DM;
  const float* bf = Bf  + (size_t)n   * DM;
  float s = 0.0f;
  for (int d = 0; d < DM; ++d) s = fmaf(e[d], bf[d], s);
  psiss[idx] = s;
}

__global__ void ass_norm(float* __restrict__ ass)
{
  int idx = blockIdx.x * blockDim.x + threadIdx.x;    // i*512 + j
  if (idx >= NN * NN) return;
  float* a = ass + (size_t)idx * KK;
  float s = a[0] + a[1] + a[2];
  float inv = 1.0f / s;
  a[0] *= inv; a[1] *= inv; a[2] *= inv;
}

__global__ void zero_f32(float* __restrict__ p, size_t count)
{
  size_t idx = (size_t)blockIdx.x * blockDim.x + threadIdx.x;
  if (idx < count) p[idx] = 0.0f;
}

__global__ void ssum_k(const float* __restrict__ mbar, float* __restrict__ Ssum)
{
  int idx = blockIdx.x * blockDim.x + threadIdx.x;    // i*7 + c
  if (idx >= NN * CC) return;
  int i = idx / CC, c = idx - i * CC;
  float s = 0.0f;
  for (int b = 0; b < NN; ++b)
    s += mbar[((size_t)b * NN + i) * CC + c];
  Ssum[idx] = s;
}

__global__ void lbp_step(const float* __restrict__ phis, const float* __restrict__ psiss,
                         const float* __restrict__ Ssum, const float* __restrict__ mbar_cur,
                         const unsigned char* __restrict__ masks,
                         float* __restrict__ mbar_next)
{
  int idx = blockIdx.x * blockDim.x + threadIdx.x;    // i*512 + j
  if (idx >= NN * NN) return;
  int i = idx >> 9;
  int j = idx & (NN - 1);

  float base[CC];
#pragma unroll
  for (int c = 0; c < CC; ++c)
    base[c] = psiss[i * CC + c] + Ssum[i * CC + c]
            - mbar_cur[((size_t)j * NN + i) * CC + c];

  float expm[CC]; float denom = 0.0f;
#pragma unroll
  for (int p = 0; p < CC; ++p) {
    float e = 0.0f;
    if (masks[j * CC + p]) {
      float mv = -INFINITY;
      const float* ph = phis + (size_t)(i * CC) * NC + (size_t)(j * CC) + p;
#pragma unroll
      for (int c = 0; c < CC; ++c)
        mv = fmaxf(mv, ph[(size_t)c * NC] + base[c]);
      e = __expf(mv);
    }
    expm[p] = e; denom += e;
  }
  float inv = 0.5f / denom;
  const float* mc = mbar_cur  + (size_t)idx * CC;
  float*       mn = mbar_next + (size_t)idx * CC;
#pragma unroll
  for (int c = 0; c < CC; ++c)
    mn[c] = __logf(0.5f * __expf(mc[c]) + expm[c] * inv);
}

__global__ void final_k(const float* __restrict__ mbar, const float* __restrict__ psiss,
                        const float* __restrict__ p_e_m, const unsigned char* __restrict__ masks,
                        const float* __restrict__ W1, const float* __restrict__ b1,
                        const float* __restrict__ W2, const float* __restrict__ b2,
                        float* __restrict__ out)
{
  int n = blockIdx.x * blockDim.x + threadIdx.x;
  if (n >= NN) return;

  float u[CC];
  for (int c = 0; c < CC; ++c) {
    float mk = masks[n * CC + c] ? 1.0f : 0.0f;
    float s = 0.0f;
    for (int i = 0; i < NN; ++i)
      if (i != n) s += mbar[((size_t)i * NN + n) * CC + c];
    u[c] = psiss[n * CC + c] + s * mk;
  }
  float den = 0.0f, ub[CC];
  for (int c = 0; c < CC; ++c) {
    float mk = masks[n * CC + c] ? 1.0f : 0.0f;
    ub[c] = __expf(u[c]) * mk; den += ub[c];
  }
  float invden = 1.0f / den;
  for (int c = 0; c < CC; ++c) {
    float g0 = ub[c] * invden;
    float g1 = p_e_m[n * CC + c];
    float p = b2[0];
    for (int jj = 0; jj < 100; ++jj) {
      float h = fmaf(g0, W1[jj * 2], fmaf(g1, W1[jj * 2 + 1], b1[jj]));
      h = h > 0.0f ? h : 0.0f;
      p = fmaf(h, W2[jj], p);
    }
    out[n * CC + c] = p;
  }
}

// ---------------------------------------------------------------------------
extern "C" void kernel_launch(void* const* d_in, const int* in_sizes, int n_in,
                              void* d_out, int out_size, void* d_ws, size_t ws_size,
                              hipStream_t stream)
{
  const float* emb     = (const float*)d_in[0];   // (512,7,300)
  const float* fmc_in  = (const float*)d_in[1];   // (512,900)
  const float* p_e_m   = (const float*)d_in[2];   // (512,7)
  const float* W_fmc   = (const float*)d_in[3];   // (300,900)
  const float* b_fmc   = (const float*)d_in[4];   // (300,)
  const float* Bmat    = (const float*)d_in[5];   // (300,300)
  const float* Rmat    = (const float*)d_in[6];   // (3,300,300) [k,e,d]
  const float* Dmat    = (const float*)d_in[7];   // (3,300,300) [k,d,e]
  const float* W_g1    = (const float*)d_in[8];   // (100,2)
  const float* b_g1    = (const float*)d_in[9];   // (100,)
  const float* W_g2    = (const float*)d_in[10];  // (1,100)
  const float* b_g2    = (const float*)d_in[11];  // (1,)
  const unsigned char* masks = (const unsigned char*)d_in[12];  // (512,7) bool

  float* ws = (float*)d_ws;
  size_t off = 0;
  float* fmcs  = ws + off; off += (size_t)NN * DM;        // 512x300
  float* Bf    = ws + off; off += (size_t)NN * DM;        // 512x300
  float* psiss = ws + off; off += (size_t)NC;             // 512x7
  float* Ssum  = ws + off; off += (size_t)NC;             // 512x7
  float* Df    = ws + off; off += (size_t)KK * NN * DM;   // 3x512x300
  float* ass   = ws + off; off += (size_t)NN * NN * KK;   // 512x512x3
  float* Wp    = ws + off; off += (size_t)DMP * 900;      // padded W_fmc  304x900
  float* Bp    = ws + off; off += (size_t)DMP * DM;       // padded B      304x300
  float* Dp    = ws + off; off += (size_t)KK * DMP * DM;  // padded D_k    3x304x300
  float* Rtp   = ws + off; off += (size_t)KK * DMP * DM;  // padded R_k^T  3x304x300
  float* val   = ws + off; off += (size_t)KK * NC * DM;   // 3x3584x300
  float* phis  = ws + off; off += (size_t)NC * NC;        // 3584x3584
  float* mbarA = ws + off; off += (size_t)NN * NN * CC;   // 512x512x7
  float* mbarB = ws + off; off += (size_t)NN * NN * CC;
  (void)ws_size; (void)in_sizes; (void)n_in; (void)out_size;

  const dim3 wave(32, 1, 1);
  const float inv_sqrt_dm = 1.0f / sqrtf((float)DM);
  const int tilesDMP = DMP / 16;   // 19

  // 0) zero-pad narrow (300-row) B-operands so the GEMM K-loop is branch-free
  pad_rows<<<(DMP * 900 + 255) / 256, 256, 0, stream>>>(W_fmc, Wp, DM, 900, DMP);
  pad_rows<<<(DMP * DM + 255) / 256, 256, 0, stream>>>(Bmat, Bp, DM, DM, DMP);
  for (int k = 0; k < KK; ++k)
    pad_rows<<<(DMP * DM + 255) / 256, 256, 0, stream>>>(
        Dmat + (size_t)k * DM * DM, Dp + (size_t)k * DMP * DM, DM, DM, DMP);
  transpose_pad_R<<<(KK * DMP * DM + 255) / 256, 256, 0, stream>>>(Rmat, Rtp);

  // 1) fmcs = tanh(fmc_in @ W_fmc^T + b_fmc)        M=512 N=300 K=900
  gemm_nt_wmma<<<dim3(tilesDMP, NN / 16), wave, 0, stream>>>(
      fmc_in, Wp, fmcs, b_fmc, NN, DM, 900, 900, 900, DM, 1, 0, 1, 0.0f);

  // 2) Bf = fmcs @ B^T                              M=512 N=300 K=300
  gemm_nt_wmma<<<dim3(tilesDMP, NN / 16), wave, 0, stream>>>(
      fmcs, Bp, Bf, nullptr, NN, DM, DM, DM, DM, DM, 1, 0, 0, 0.0f);

  // 3) psiss[n,c] = emb[n,c,:] . Bf[n,:]
  psiss_k<<<(NC + 255) / 256, 256, 0, stream>>>(emb, Bf, psiss);

  // 4) Df_k = fmcs @ D_k^T                          M=512 N=300 K=300
  for (int k = 0; k < KK; ++k)
    gemm_nt_wmma<<<dim3(tilesDMP, NN / 16), wave, 0, stream>>>(
        fmcs, Dp + (size_t)k * DMP * DM, Df + (size_t)k * NN * DM, nullptr,
        NN, DM, DM, DM, DM, DM, 1, 0, 0, 0.0f);

  // 5) x_k[i,j] = exp((fmcs @ Df_k^T)/sqrt(DM)) stored interleaved: ass[(i*512+j)*3+k]
  for (int k = 0; k < KK; ++k)
    gemm_nt_wmma<<<dim3(NN / 16, NN / 16), wave, 0, stream>>>(
        fmcs, Df + (size_t)k * NN * DM, ass, nullptr,
        NN, NN, DM, DM, DM, NN * KK, KK, k, 2, inv_sqrt_dm);

  // 6) normalize over k
  ass_norm<<<(NN * NN + 255) / 256, 256, 0, stream>>>(ass);

  // 7) val_k = emb @ Rt_k^T                         M=3584 N=300 K=300
  for (int k = 0; k < KK; ++k)
    gemm_nt_wmma<<<dim3(tilesDMP, NC / 16), wave, 0, stream>>>(
        emb, Rtp + (size_t)k * DMP * DM, val + (size_t)k * NC * DM, nullptr,
        NC, DM, DM, DM, DM, DM, 1, 0, 0, 0.0f);

  // 8) phis2[(i,c),(j,p)] = sum_k ass[i,j,k] * (val_k @ emb^T)  3x(3584x3584x300) fused
  phis_wmma<<<dim3(NC / 16, NC / 16), wave, 0, stream>>>(val, emb, ass, phis);

  // 9) LBP loop, ping-pong mbar buffers
  zero_f32<<<((size_t)NN * NN * CC + 255) / 256, 256, 0, stream>>>(mbarA, (size_t)NN * NN * CC);
  float* cur = mbarA;
  float* nxt = mbarB;
  for (int it = 0; it < LBP; ++it) {
    ssum_k<<<(NC + 255) / 256, 256, 0, stream>>>(cur, Ssum);
    lbp_step<<<(NN * NN + 255) / 256, 256, 0, stream>>>(phis, psiss, Ssum, cur, masks, nxt);
    float* t = cur; cur = nxt; nxt = t;
  }

  // 10) u -> ubar -> 2-layer MLP -> p
  final_k<<<(NN + 255) / 256, 256, 0, stream>>>(
      cur, psiss, p_e_m, masks, W_g1, b_g1, W_g2, b_g2, (float*)d_out);
}